// BaseModule_71451075936278
// MI455X (gfx1250) — compile-verified
//
#include <hip/hip_runtime.h>
#include <hip/hip_bf16.h>
#include <math.h>

// ---------------------------------------------------------------------------
// GAT-style layer for MI455X (gfx1250, wave32).
//   z = h @ W_fc^T           -> V_WMMA_F32_16X16X4_F32 (fp32 matrix core)
//   s1/s2 per-node scalars   -> turns z[dst] vector gather into scalar gather
//   3 edge passes            -> segment softmax + weighted scatter (L2-resident)
// ---------------------------------------------------------------------------

typedef __attribute__((ext_vector_type(2))) float v2f;
typedef __attribute__((ext_vector_type(8))) float v8f;

#define DWIDTH 64   // D_IN == D_OUT == 64

// ---- float atomic max via int/uint atomics (lowers to native global atomics)
__device__ __forceinline__ void atomicMaxFloat(float* addr, float val) {
    if (val >= 0.0f) {
        atomicMax((int*)addr, __float_as_int(val));
    } else {
        atomicMin((unsigned int*)addr, __float_as_uint(val));
    }
}

// ---- init: out = 0, m = -inf, denom = 0 -----------------------------------
__global__ void init_kernel(float* __restrict__ out, float* __restrict__ m,
                            float* __restrict__ denom, int N) {
    int i = blockIdx.x * blockDim.x + threadIdx.x;
    if (i < N * DWIDTH) out[i] = 0.0f;
    if (i < N) { m[i] = -INFINITY; denom[i] = 0.0f; }
}

// ---- z = h @ W_fc^T via V_WMMA_F32_16X16X4_F32 ----------------------------
// One wave computes a 16x16 tile of z. 16 K-steps of K=4 accumulate in v8f.
// A (16x4 f32, 2 VGPRs): lanes 0-15 -> M=lane, K={k0,k0+1};
//                        lanes 16-31 -> M=lane-16, K={k0+2,k0+3}.
// B (4x16 f32, 2 VGPRs): row-striped across lanes (symmetric layout):
//                        lanes 0-15 -> K={k0,k0+1}, N=lane;
//                        lanes 16-31 -> K={k0+2,k0+3}, N=lane-16.
// B[k][n] = W_fc[n][k] (row n of W_fc), so both A and B load contiguous pairs.
__global__ __launch_bounds__(256) void gemm_z_wmma(const float* __restrict__ h,
                                                   const float* __restrict__ Wfc,
                                                   float* __restrict__ z, int N) {
    const int wave = (blockIdx.x * blockDim.x + threadIdx.x) >> 5;
    const int lane = threadIdx.x & 31;
    const int colTiles = DWIDTH / 16;                 // 4
    const int rowTile  = wave / colTiles;
    const int colTile  = wave % colTiles;
    const int row0 = rowTile * 16;
    if (row0 >= N) return;                            // wave-uniform: EXEC stays all-1s
    const int o0   = colTile * 16;
    const int half = lane >> 4;                       // 0 | 1
    const int l16  = lane & 15;

    const float* __restrict__ arow = h   + (size_t)(row0 + l16) * DWIDTH;
    const float* __restrict__ brow = Wfc + (size_t)(o0   + l16) * DWIDTH;

    v8f c = {};
#pragma unroll
    for (int k0 = 0; k0 < DWIDTH; k0 += 4) {
        const int ka = k0 + half * 2;
        v2f a, b;
        a.x = arow[ka]; a.y = arow[ka + 1];
        b.x = brow[ka]; b.y = brow[ka + 1];
        c = __builtin_amdgcn_wmma_f32_16x16x4_f32(
                /*neg_a=*/false, a, /*neg_b=*/false, b,
                /*c_mod=*/(short)0, c, /*reuse_a=*/false, /*reuse_b=*/false);
    }

    // C/D layout: VGPR j holds (M = j + 8*half, N = l16)
    float* __restrict__ zt = z + (size_t)(row0 + 8 * half) * DWIDTH + o0 + l16;
#pragma unroll
    for (int j = 0; j < 8; ++j)
        zt[(size_t)j * DWIDTH] = c[j];
}

// ---- per-node attention scalars: s1=<z[v],wa[0:64]>, s2=<z[v],wa[64:128]> --
__global__ __launch_bounds__(256) void node_scalars(const float* __restrict__ z,
                                                    const float* __restrict__ Wattn,
                                                    float* __restrict__ s1,
                                                    float* __restrict__ s2, int N) {
    const int wave = (blockIdx.x * blockDim.x + threadIdx.x) >> 5;
    const int lane = threadIdx.x & 31;
    if (wave >= N) return;
    const float* __restrict__ zr = z + (size_t)wave * DWIDTH;
    float a1 = zr[lane] * Wattn[lane]      + zr[lane + 32] * Wattn[lane + 32];
    float a2 = zr[lane] * Wattn[64 + lane] + zr[lane + 32] * Wattn[96 + lane];
#pragma unroll
    for (int off = 16; off > 0; off >>= 1) {
        a1 += __shfl_down(a1, off);
        a2 += __shfl_down(a2, off);
    }
    if (lane == 0) { s1[wave] = a1; s2[wave] = a2; }
}

// ---- edge pass 1: score + leaky_relu + segment max -------------------------
__global__ void edge_pass1(const float* __restrict__ e,
                           const int* __restrict__ src, const int* __restrict__ dst,
                           const float* __restrict__ s1, const float* __restrict__ s2,
                           const float* __restrict__ Wedge, const float* __restrict__ Wattn,
                           float* __restrict__ a_out, float* __restrict__ m, int E) {
    int i = blockIdx.x * blockDim.x + threadIdx.x;
    if (i >= E) return;
    const float e0 = e[2 * i], e1 = e[2 * i + 1];
    const float ex0 = e0 * Wedge[0] + e1 * Wedge[1];   // W_edge row 0
    const float ex1 = e0 * Wedge[2] + e1 * Wedge[3];   // W_edge row 1
    float a = s1[src[i]] + s2[dst[i]] + ex0 * Wattn[128] + ex1 * Wattn[129];
    a = (a > 0.0f) ? a : 0.01f * a;                    // leaky_relu
    a_out[i] = a;
    atomicMaxFloat(&m[dst[i]], a);
}

// ---- edge pass 2: ea = exp(a - m[dst]); denom += ea ------------------------
__global__ void edge_pass2(const int* __restrict__ dst,
                           const float* __restrict__ a_in,
                           const float* __restrict__ m,
                           float* __restrict__ ea, float* __restrict__ denom, int E) {
    int i = blockIdx.x * blockDim.x + threadIdx.x;
    if (i >= E) return;
    const int d = dst[i];
    const float v = expf(a_in[i] - m[d]);
    ea[i] = v;
    atomicAdd(&denom[d], v);
}

// ---- edge pass 3: out[dst] += attn * (z[src] + ez) -------------------------
// One wave per edge; each lane handles features {lane, lane+32}.
__global__ __launch_bounds__(256) void edge_pass3(const float* __restrict__ e,
                                                  const int* __restrict__ src,
                                                  const int* __restrict__ dst,
                                                  const float* __restrict__ ea,
                                                  const float* __restrict__ denom,
                                                  const float* __restrict__ z,
                                                  const float* __restrict__ Wedge,
                                                  const float* __restrict__ Wez,
                                                  float* __restrict__ out, int E) {
    const int wave = (blockIdx.x * blockDim.x + threadIdx.x) >> 5;
    const int lane = threadIdx.x & 31;
    if (wave >= E) return;
    const int s = src[wave], d = dst[wave];
    const float attn = ea[wave] / denom[d];
    const float e0 = e[2 * wave], e1 = e[2 * wave + 1];
    const float ex0 = e0 * Wedge[0] + e1 * Wedge[1];
    const float ex1 = e0 * Wedge[2] + e1 * Wedge[3];
    const float* __restrict__ zs = z + (size_t)s * DWIDTH;
    float* __restrict__ o = out + (size_t)d * DWIDTH;
#pragma unroll
    for (int k = lane; k < DWIDTH; k += 32) {
        const float ez = ex0 * Wez[2 * k] + ex1 * Wez[2 * k + 1];
        atomicAdd(&o[k], attn * (zs[k] + ez));
    }
}

// ---------------------------------------------------------------------------
extern "C" void kernel_launch(void* const* d_in, const int* in_sizes, int n_in,
                              void* d_out, int out_size, void* d_ws, size_t ws_size,
                              hipStream_t stream) {
    const float* h     = (const float*)d_in[0];
    const float* e     = (const float*)d_in[1];
    const float* Wfc   = (const float*)d_in[2];
    const float* Wattn = (const float*)d_in[3];
    const float* Wedge = (const float*)d_in[4];
    const float* Wez   = (const float*)d_in[5];
    const int*   src   = (const int*)d_in[6];
    const int*   dst   = (const int*)d_in[7];
    float* out = (float*)d_out;

    const int N = in_sizes[0] / DWIDTH;   // 100000
    const int E = in_sizes[6];            // 1600000

    // workspace layout (floats)
    float* ws    = (float*)d_ws;
    float* z     = ws;                       // N*64
    float* s1    = z + (size_t)N * DWIDTH;   // N
    float* s2    = s1 + N;                   // N
    float* m     = s2 + N;                   // N
    float* denom = m + N;                    // N
    float* a     = denom + N;                // E
    float* ea    = a + E;                    // E

    const int B = 256;

    // 1. init out / m / denom
    init_kernel<<<(N * DWIDTH + B - 1) / B, B, 0, stream>>>(out, m, denom, N);

    // 2. z = h @ W_fc^T via WMMA (waves = (N/16) * 4)
    {
        const int waves = (N / 16) * (DWIDTH / 16);
        const int blocks = (waves * 32 + B - 1) / B;
        gemm_z_wmma<<<blocks, B, 0, stream>>>(h, Wfc, z, N);
    }

    // 3. per-node attention scalars (one wave per node)
    node_scalars<<<((size_t)N * 32 + B - 1) / B, B, 0, stream>>>(z, Wattn, s1, s2, N);

    // 4. edge pass 1: score + segment max
    edge_pass1<<<(E + B - 1) / B, B, 0, stream>>>(e, src, dst, s1, s2, Wedge, Wattn, a, m, E);

    // 5. edge pass 2: exp + segment sum
    edge_pass2<<<(E + B - 1) / B, B, 0, stream>>>(dst, a, m, ea, denom, E);

    // 6. edge pass 3: weighted scatter-add (one wave per edge)
    edge_pass3<<<((size_t)E * 32 + B - 1) / B, B, 0, stream>>>(e, src, dst, ea, denom, z,
                                                              Wedge, Wez, out, E);
}